// SGC_10196252361440
// MI455X (gfx1250) — compile-verified
//
#include <hip/hip_runtime.h>

typedef float v2f __attribute__((ext_vector_type(2)));
typedef float v8f __attribute__((ext_vector_type(8)));

constexpr int FEAT = 256;   // IN_FEATS == HIDDEN
constexpr int NCLS = 64;

// ---------------------------------------------------------------- utilities
__global__ void sgc_zero_f32(float* __restrict__ p, long n) {
  long i = (long)blockIdx.x * blockDim.x + threadIdx.x;
  if (i < n) p[i] = 0.0f;
}

__global__ void sgc_degree(const int* __restrict__ dst, float* __restrict__ deg, int ne) {
  int e = blockIdx.x * blockDim.x + threadIdx.x;
  if (e < ne) unsafeAtomicAdd(&deg[dst[e]], 1.0f);
}

__global__ void sgc_make_norm(float* __restrict__ dn, int n) {
  int i = blockIdx.x * blockDim.x + threadIdx.x;
  if (i < n) dn[i] = rsqrtf(fmaxf(dn[i], 1.0f));
}

// ------------------------------------------- LayerNorm fused with pre-scale
// out[row] = ((x - mu) * rsqrt(var+eps) * gamma + beta) * norm[row]
// one wave32 per row, 8 floats per lane via float4 x2, wave reduction
__global__ __launch_bounds__(256) void sgc_ln(const float* __restrict__ x,
    const float* __restrict__ gamma, const float* __restrict__ beta,
    const float* __restrict__ norm, float* __restrict__ out, int n) {
  int wave = threadIdx.x >> 5, lane = threadIdx.x & 31;
  int row = blockIdx.x * 8 + wave;
  if (row >= n) return;
  const float4* xr = (const float4*)(x + (size_t)row * FEAT);
  float4 v0 = xr[lane];
  float4 v1 = xr[lane + 32];
  float s  = v0.x + v0.y + v0.z + v0.w + v1.x + v1.y + v1.z + v1.w;
  float ss = v0.x*v0.x + v0.y*v0.y + v0.z*v0.z + v0.w*v0.w
           + v1.x*v1.x + v1.y*v1.y + v1.z*v1.z + v1.w*v1.w;
  #pragma unroll
  for (int off = 16; off > 0; off >>= 1) {
    s  += __shfl_xor(s, off);
    ss += __shfl_xor(ss, off);
  }
  float mu  = s * (1.0f / FEAT);
  float var = ss * (1.0f / FEAT) - mu * mu;
  float rs  = rsqrtf(var + 1e-5f);
  float nr  = norm[row];
  const float4* g4 = (const float4*)gamma;
  const float4* b4 = (const float4*)beta;
  float4* orow = (float4*)(out + (size_t)row * FEAT);
  float4 ga = g4[lane], be = b4[lane], o;
  o.x = ((v0.x - mu) * rs * ga.x + be.x) * nr;
  o.y = ((v0.y - mu) * rs * ga.y + be.y) * nr;
  o.z = ((v0.z - mu) * rs * ga.z + be.z) * nr;
  o.w = ((v0.w - mu) * rs * ga.w + be.w) * nr;
  orow[lane] = o;
  ga = g4[lane + 32]; be = b4[lane + 32];
  o.x = ((v1.x - mu) * rs * ga.x + be.x) * nr;
  o.y = ((v1.y - mu) * rs * ga.y + be.y) * nr;
  o.z = ((v1.z - mu) * rs * ga.z + be.z) * nr;
  o.w = ((v1.w - mu) * rs * ga.w + be.w) * nr;
  orow[lane + 32] = o;
}

// ------------------------------------------------- edge scatter-accumulate
// out[dst] += g[src] * (use_norm2 ? norm[src]^2 : 1)
// one wave32 per edge; lane-contiguous loads + fadd atomics (coalesced)
__global__ __launch_bounds__(256) void sgc_scatter(const float* __restrict__ g,
    const int* __restrict__ esrc, const int* __restrict__ edst,
    const float* __restrict__ norm, float* __restrict__ out, int ne, int use_norm2) {
  int wave = threadIdx.x >> 5, lane = threadIdx.x & 31;
  int e = blockIdx.x * 8 + wave;
  if (e >= ne) return;
  int s = esrc[e], d = edst[e];
  float scale = 1.0f;
  if (use_norm2) { float nv = norm[s]; scale = nv * nv; }
  const float* grow = g + (size_t)s * FEAT;
  float* orow = out + (size_t)d * FEAT;
  #pragma unroll
  for (int c = 0; c < 8; ++c) {
    int f = lane + 32 * c;
    unsafeAtomicAdd(orow + f, grow[f] * scale);
  }
}

// --------------------------------------- fused dual-GEMM via V_WMMA f32 K=4
// stage1: hc = relu((S3*norm) @ Wc^T + bc)   [16 x 256]  (LDS)
// stage2: out = hc @ Wf^T + bf               [16 x 64]
// block = 128 threads (4 waves), 16 rows per block
__global__ __launch_bounds__(128) void sgc_gemm_fused(const float* __restrict__ h,
    const float* __restrict__ norm,
    const float* __restrict__ Wc, const float* __restrict__ bc,
    const float* __restrict__ Wf, const float* __restrict__ bf,
    float* __restrict__ out, int n) {
  constexpr int LDW = FEAT + 4;                  // pad: 260 % 64 = 4 -> no bank conflicts
  __shared__ float shA[16 * LDW];
  __shared__ float shC[16 * LDW];
  int tid = threadIdx.x;
  int m0 = blockIdx.x * 16;

  // cooperative load of 16x256 A-tile, fused final norm scaling
  for (int t = tid; t < 16 * (FEAT / 4); t += 128) {
    int row = t >> 6;                            // FEAT/4 == 64
    int c4  = t & 63;
    int gr  = m0 + row;
    float4 v = make_float4(0.f, 0.f, 0.f, 0.f);
    float nr = 0.f;
    if (gr < n) {
      v  = ((const float4*)(h + (size_t)gr * FEAT))[c4];
      nr = norm[gr];
    }
    float* dp = &shA[row * LDW + c4 * 4];
    dp[0] = v.x * nr; dp[1] = v.y * nr; dp[2] = v.z * nr; dp[3] = v.w * nr;
  }
  __syncthreads();

  int wave = tid >> 5, lane = tid & 31;
  int mrow = lane & 15;                          // A: M = lane%16
  int kk   = (lane >> 4) << 1;                   // A: K offset 0 (lanes 0-15) / 2 (16-31)

  // ---- stage 1: conv GEMM, each wave owns 64 output columns (4 N-tiles)
  for (int nt = 0; nt < 4; ++nt) {
    int ncol = wave * 64 + nt * 16 + (lane & 15);
    const float* wrow = Wc + (size_t)ncol * FEAT;     // B[k][n] = Wc[n][k]
    v8f acc = {};
    for (int k = 0; k < FEAT; k += 4) {
      v2f a = *(const v2f*)&shA[mrow * LDW + k + kk];
      v2f b = *(const v2f*)&wrow[k + kk];
      acc = __builtin_amdgcn_wmma_f32_16x16x4_f32(false, a, false, b,
                                                  (short)0, acc, false, false);
    }
    float bias = bc[ncol];
    #pragma unroll
    for (int r = 0; r < 8; ++r) {
      int row = (lane < 16) ? r : (r + 8);            // C/D layout
      float vv = acc[r] + bias;
      shC[row * LDW + ncol] = vv > 0.f ? vv : 0.f;    // ReLU
    }
  }
  __syncthreads();

  // ---- stage 2: classifier GEMM, each wave owns one 16-column N-tile
  {
    int ncol = wave * 16 + (lane & 15);               // 0..63
    const float* wrow = Wf + (size_t)ncol * FEAT;     // B[k][n] = Wf[n][k]
    v8f acc = {};
    for (int k = 0; k < FEAT; k += 4) {
      v2f a = *(const v2f*)&shC[mrow * LDW + k + kk];
      v2f b = *(const v2f*)&wrow[k + kk];
      acc = __builtin_amdgcn_wmma_f32_16x16x4_f32(false, a, false, b,
                                                  (short)0, acc, false, false);
    }
    float bias = bf[ncol];
    #pragma unroll
    for (int r = 0; r < 8; ++r) {
      int row = (lane < 16) ? r : (r + 8);
      int gr = m0 + row;
      if (gr < n) out[(size_t)gr * NCLS + ncol] = acc[r] + bias;
    }
  }
}

// ----------------------------------------------------------------- launcher
extern "C" void kernel_launch(void* const* d_in, const int* in_sizes, int n_in,
                              void* d_out, int out_size, void* d_ws, size_t ws_size,
                              hipStream_t stream) {
  const float* features = (const float*)d_in[0];
  const int*   esrc     = (const int*)d_in[1];
  const int*   edst     = (const int*)d_in[2];
  const float* gamma    = (const float*)d_in[3];
  const float* beta     = (const float*)d_in[4];
  const float* Wc       = (const float*)d_in[5];
  const float* bc       = (const float*)d_in[6];
  const float* Wf       = (const float*)d_in[7];
  const float* bf       = (const float*)d_in[8];
  float* out = (float*)d_out;

  int n  = in_sizes[0] / FEAT;   // 100000
  int ne = in_sizes[1];          // 3200000

  // workspace: norm[n] | bufA[n*256] | bufB[n*256]   (~205 MB)
  float* wsf  = (float*)d_ws;
  size_t nAl  = ((size_t)n + 127) & ~(size_t)127;
  float* norm = wsf;
  float* bufA = wsf + nAl;
  float* bufB = bufA + (size_t)n * FEAT;

  long nf = (long)n * FEAT;

  // degrees -> norm = rsqrt(clamp(deg,1))
  sgc_zero_f32 <<<dim3((unsigned)((n + 255) / 256)), dim3(256), 0, stream>>>(norm, n);
  sgc_degree   <<<dim3((unsigned)((ne + 255) / 256)), dim3(256), 0, stream>>>(edst, norm, ne);
  sgc_make_norm<<<dim3((unsigned)((n + 255) / 256)), dim3(256), 0, stream>>>(norm, n);

  // bufA = LN(x) * norm    (pre-scaled for hop 1)
  sgc_ln<<<dim3((unsigned)((n + 7) / 8)), dim3(256), 0, stream>>>(
      features, gamma, beta, norm, bufA, n);

  // three hops of S <- scatter(S_prev * norm^2[src]); scaling folded into load
  for (int hop = 0; hop < 3; ++hop) {
    sgc_zero_f32<<<dim3((unsigned)((nf + 255) / 256)), dim3(256), 0, stream>>>(bufB, nf);
    sgc_scatter <<<dim3((unsigned)((ne + 7) / 8)), dim3(256), 0, stream>>>(
        bufA, esrc, edst, norm, bufB, ne, hop > 0 ? 1 : 0);
    float* t = bufA; bufA = bufB; bufB = t;
  }

  // fused (S3*norm) @ Wc^T -> relu -> @ Wf^T, WMMA f32
  sgc_gemm_fused<<<dim3((unsigned)((n + 15) / 16)), dim3(128), 0, stream>>>(
      bufA, norm, Wc, bc, Wf, bf, out, n);
}